// Graph_45011257262603
// MI455X (gfx1250) — compile-verified
//
#include <hip/hip_runtime.h>
#include <hip/hip_bf16.h>

typedef __attribute__((ext_vector_type(2))) float v2f;
typedef __attribute__((ext_vector_type(8))) float v8f;

#define D 128            // D_IN == D_OUT == 128
#define LDS_STRIDE 132   // 128 + 4 floats pad -> stride 132 mod 64 = 4 -> conflict-free b64 LDS reads

// ---------------------------------------------------------------------------
// Kernel 1: zero the aggregation workspace (float4 stores)
// ---------------------------------------------------------------------------
__global__ __launch_bounds__(256) void gc_zero(float4* __restrict__ p, long long n4) {
    long long i = (long long)blockIdx.x * blockDim.x + threadIdx.x;
    if (i < n4) {
        float4 z; z.x = 0.f; z.y = 0.f; z.z = 0.f; z.w = 0.f;
        p[i] = z;
    }
}

// ---------------------------------------------------------------------------
// Kernel 2: edge scatter  agg[dst] += w * x[src]
// 32 lanes cooperate on one edge (lane -> 4-float feature chunk), so the
// int64 index loads are wave-uniform (scalar-broadcast friendly) and the
// x gather is a coalesced b128 per lane. Atomics land in L2 (agg fits).
// ---------------------------------------------------------------------------
__global__ __launch_bounds__(256) void gc_scatter(
    const float* __restrict__ x,
    const long long* __restrict__ src,
    const long long* __restrict__ dst,
    const float* __restrict__ ew,
    float* __restrict__ agg,
    long long n_edges)
{
    long long gid = (long long)blockIdx.x * blockDim.x + threadIdx.x;
    long long e = gid >> 5;                 // 32 work-items per edge
    if (e >= n_edges) return;
    int seg = (int)(gid & 31);              // feature chunk: 4 floats

    long long s = src[e];
    long long d = dst[e];
    float w = ew[e];

    const float4 v = *(const float4*)(x + s * D + seg * 4);
    float* a = agg + d * D + seg * 4;
    atomicAdd(a + 0, v.x * w);
    atomicAdd(a + 1, v.y * w);
    atomicAdd(a + 2, v.z * w);
    atomicAdd(a + 3, v.w * w);
}

// ---------------------------------------------------------------------------
// Kernel 3: fused dual GEMM  out = agg @ Wrel^T + x @ Wroot^T + bias
// One block = one 16-row node tile; 8 waves = 8 col-tiles of 16 features.
// Tiles staged in LDS; K-loop of V_WMMA_F32_16X16X4_F32 (f32 end to end).
// ---------------------------------------------------------------------------
__global__ __launch_bounds__(256) void gc_gemm(
    const float* __restrict__ agg,
    const float* __restrict__ x,
    const float* __restrict__ Wrel,    // [D_OUT][D_IN] row-major
    const float* __restrict__ Wroot,   // [D_OUT][D_IN] row-major
    const float* __restrict__ bias,    // [D_OUT]
    float* __restrict__ out,           // [N][D_OUT]
    long long n_nodes)
{
    __shared__ float s_agg[16 * LDS_STRIDE];
    __shared__ float s_x[16 * LDS_STRIDE];

    const long long row0 = (long long)blockIdx.x * 16;
    const int tid = threadIdx.x;

    // Cooperative tile load: 16 rows x 128 cols = 512 float4; 2 per thread.
    #pragma unroll
    for (int i = tid; i < 512; i += 256) {
        int r = i >> 5;                  // 32 float4 per row
        int c = (i & 31) << 2;           // float column
        long long gr = row0 + r;
        if (gr >= n_nodes) gr = n_nodes - 1;   // clamp (N is a multiple of 16 anyway)
        float4 va = *(const float4*)(agg + gr * D + c);
        float4 vx = *(const float4*)(x   + gr * D + c);
        *(float4*)(s_agg + r * LDS_STRIDE + c) = va;
        *(float4*)(s_x   + r * LDS_STRIDE + c) = vx;
    }
    __syncthreads();

    const int wave = tid >> 5;           // 0..7 -> 16-wide output column tile
    const int lane = tid & 31;
    const int lo16 = lane & 15;          // A: row M ; B: col N ; D: col N
    const int hi   = lane >> 4;          // selects K pair / row half

    const int col0 = wave * 16;
    const float* wrel_row  = Wrel  + (size_t)(col0 + lo16) * D;  // B[k][n] = W[n][k]
    const float* wroot_row = Wroot + (size_t)(col0 + lo16) * D;
    const float* sa = s_agg + lo16 * LDS_STRIDE;
    const float* sx = s_x   + lo16 * LDS_STRIDE;

    v8f acc = {};
    #pragma unroll
    for (int k0 = 0; k0 < D; k0 += 4) {
        const int k = k0 + 2 * hi;       // ISA: VGPR j holds K = 2*hi + j (A and B alike)
        v2f a_agg  = { sa[k],          sa[k + 1] };
        v2f a_x    = { sx[k],          sx[k + 1] };
        v2f b_rel  = { wrel_row[k],    wrel_row[k + 1] };
        v2f b_root = { wroot_row[k],   wroot_row[k + 1] };
        // (neg_a, A, neg_b, B, c_mod, C, reuse_a, reuse_b)
        acc = __builtin_amdgcn_wmma_f32_16x16x4_f32(
                  false, a_agg, false, b_rel, (short)0, acc, false, false);
        acc = __builtin_amdgcn_wmma_f32_16x16x4_f32(
                  false, a_x,   false, b_root, (short)0, acc, false, false);
    }

    const float bv = bias[col0 + lo16];
    // D layout: VGPR r -> row M = r + 8*hi, col N = lane&15
    #pragma unroll
    for (int r = 0; r < 8; ++r) {
        long long m = row0 + r + 8 * hi;
        if (m < n_nodes)
            out[m * D + col0 + lo16] = acc[r] + bv;
    }
}

// ---------------------------------------------------------------------------
extern "C" void kernel_launch(void* const* d_in, const int* in_sizes, int n_in,
                              void* d_out, int out_size, void* d_ws, size_t ws_size,
                              hipStream_t stream) {
    const float*     x     = (const float*)d_in[0];
    const long long* eidx  = (const long long*)d_in[1];   // int64 [2, E]
    const float*     ew    = (const float*)d_in[2];
    const float*     Wrel  = (const float*)d_in[3];
    const float*     Wroot = (const float*)d_in[4];
    const float*     bias  = (const float*)d_in[5];
    float*           out   = (float*)d_out;
    float*           agg   = (float*)d_ws;                // N*D floats = 51.2 MB

    const long long nN = (long long)in_sizes[0] / D;      // 100000
    const long long nE = (long long)in_sizes[2];          // 640000
    const long long* src = eidx;                          // edge_index[0]
    const long long* dst = eidx + nE;                     // edge_index[1]

    // 1) zero agg
    {
        long long n4 = nN * D / 4;
        int blocks = (int)((n4 + 255) / 256);
        gc_zero<<<blocks, 256, 0, stream>>>((float4*)agg, n4);
    }
    // 2) scatter edges
    {
        long long work = nE * 32;
        int blocks = (int)((work + 255) / 256);
        gc_scatter<<<blocks, 256, 0, stream>>>(x, src, dst, ew, agg, nE);
    }
    // 3) fused dual GEMM with WMMA f32
    {
        int blocks = (int)((nN + 15) / 16);               // 6250
        gc_gemm<<<blocks, 256, 0, stream>>>(agg, x, Wrel, Wroot, bias, out, nN);
    }
}